// GCNCat_2860448219407
// MI455X (gfx1250) — compile-verified
//
#include <hip/hip_runtime.h>
#include <hip/hip_bf16.h>

typedef float v2f __attribute__((ext_vector_type(2)));
typedef float v8f __attribute__((ext_vector_type(8)));

#define NODES 50000
#define EDGES 800000
#define NGRAPH 32
#define NCLASS 10
#define LDH 208

// ---------------------------------------------------------------------------
// Build inp = [norm|pos|x] into H208 columns 64..71 and 200..207
// ---------------------------------------------------------------------------
__global__ void pack_inp_kernel(const float* __restrict__ norm,
                                const float* __restrict__ pos,
                                const float* __restrict__ x,
                                float* __restrict__ H) {
    int idx = blockIdx.x * blockDim.x + threadIdx.x;
    if (idx >= NODES * 8) return;
    int row = idx >> 3;
    int c = idx & 7;
    float v;
    if (c < 3)      v = norm[row * 3 + c];
    else if (c < 6) v = pos[row * 3 + (c - 3)];
    else            v = x[row * 2 + (c - 6)];
    H[row * LDH + 64 + c]  = v;
    H[row * LDH + 200 + c] = v;
}

// ---------------------------------------------------------------------------
// Zero an F-wide column block of a strided matrix
// ---------------------------------------------------------------------------
template <int F>
__global__ void zero_block_kernel(float* __restrict__ dst, int ld, int c0) {
    int idx = blockIdx.x * blockDim.x + threadIdx.x;
    if (idx >= NODES * F) return;
    int row = idx / F;
    int f = idx % F;
    dst[row * ld + c0 + f] = 0.0f;
}

__global__ void zero_u32_kernel(unsigned int* __restrict__ p, int n) {
    int idx = blockIdx.x * blockDim.x + threadIdx.x;
    if (idx < n) p[idx] = 0u;
}

// ---------------------------------------------------------------------------
// Dense transform m = A @ W using V_WMMA_F32_16X16X4_F32 (f32 in, f32 acc).
// One wave = one 16x16 output tile. EXEC is all ones (block = 32 threads).
// A fragment (16x4): lanes 0-15 hold K=k0+0 (v0) / k0+1 (v1) for M=lane,
//                    lanes 16-31 hold K=k0+2 / k0+3 for M=lane-16.
// B fragment (4x16): mirrored layout with N=lane&15.
// C/D (16x16, 8 VGPRs): VGPR v -> M=v (lanes 0-15) / M=v+8 (lanes 16-31).
// ---------------------------------------------------------------------------
template <int K, int LDA, int F>
__global__ void gemm_wmma_kernel(const float* __restrict__ A,
                                 const float* __restrict__ W,
                                 float* __restrict__ out) {
    int lane = threadIdx.x;          // 0..31, full wave
    int half = lane >> 4;            // 0 or 1
    int l = lane & 15;
    int tile_n = blockIdx.x;         // F/16 tiles
    int tile_m = blockIdx.y;         // NODES/16 tiles

    const float* arow = A + (size_t)(tile_m * 16 + l) * LDA;
    int coln = tile_n * 16 + l;

    v8f c = {};
#pragma unroll
    for (int k0 = 0; k0 < K; k0 += 4) {
        int ka = k0 + 2 * half;
        v2f a, b;
        a.x = arow[ka + 0];
        a.y = arow[ka + 1];
        b.x = W[(ka + 0) * F + coln];
        b.y = W[(ka + 1) * F + coln];
        c = __builtin_amdgcn_wmma_f32_16x16x4_f32(
                /*neg_a=*/false, a, /*neg_b=*/false, b,
                /*c_mod=*/(short)0, c, /*reuse_a=*/false, /*reuse_b=*/false);
    }

    int base_row = tile_m * 16 + 8 * half;
#pragma unroll
    for (int v = 0; v < 8; ++v) {
        out[(size_t)(base_row + v) * F + tile_n * 16 + l] = c[v];
    }
}

// ---------------------------------------------------------------------------
// Edge scatter: agg[dst[e], f] += m[src[e], f]
// ---------------------------------------------------------------------------
template <int F>
__global__ void scatter_add_kernel(const float* __restrict__ m,
                                   const long long* __restrict__ src,
                                   const long long* __restrict__ dst,
                                   float* __restrict__ agg,
                                   int ld_agg, int c0) {
    long long idx = (long long)blockIdx.x * blockDim.x + threadIdx.x;
    if (idx >= (long long)EDGES * F) return;
    int e = (int)(idx / F);
    int f = (int)(idx % F);
    int s = (int)src[e];
    int d = (int)dst[e];
    float v = m[(size_t)s * F + f];
    atomicAdd(&agg[(size_t)d * ld_agg + c0 + f], v);
}

// ---------------------------------------------------------------------------
// In-place bias + ReLU on a column block
// ---------------------------------------------------------------------------
template <int F>
__global__ void bias_relu_kernel(float* __restrict__ h, int ld, int c0,
                                 const float* __restrict__ b) {
    int idx = blockIdx.x * blockDim.x + threadIdx.x;
    if (idx >= NODES * F) return;
    int row = idx / F;
    int f = idx % F;
    float v = h[(size_t)row * ld + c0 + f] + b[f];
    h[(size_t)row * ld + c0 + f] = v > 0.0f ? v : 0.0f;
}

// ---------------------------------------------------------------------------
// Global max pool: pooled[batch[i], f] = max(h3[i, f]); values >= 0 after
// ReLU, so unsigned bit-pattern atomicMax == float max, and 0-init is exact.
// ---------------------------------------------------------------------------
__global__ void pool_max_kernel(const float* __restrict__ h3,
                                const long long* __restrict__ batch,
                                unsigned int* __restrict__ pooled) {
    int idx = blockIdx.x * blockDim.x + threadIdx.x;
    if (idx >= NODES * 256) return;
    int row = idx >> 8;
    int f = idx & 255;
    float v = h3[(size_t)row * 256 + f];
    int g = (int)batch[row];
    atomicMax(&pooled[g * 256 + f], __float_as_uint(v));
}

// ---------------------------------------------------------------------------
// Head: logits = pooled @ Wl + bl; out = log_softmax; pred = exp(out).
// One thread per graph (32 threads, 2560 MACs each).
// ---------------------------------------------------------------------------
__global__ void head_kernel(const unsigned int* __restrict__ pooled,
                            const float* __restrict__ Wl,
                            const float* __restrict__ bl,
                            float* __restrict__ out) {
    int g = threadIdx.x;
    if (g >= NGRAPH) return;
    float logit[NCLASS];
#pragma unroll
    for (int c = 0; c < NCLASS; ++c) logit[c] = bl[c];
    for (int f = 0; f < 256; ++f) {
        float p = __uint_as_float(pooled[g * 256 + f]);
#pragma unroll
        for (int c = 0; c < NCLASS; ++c) logit[c] += p * Wl[f * NCLASS + c];
    }
    float mx = logit[0];
#pragma unroll
    for (int c = 1; c < NCLASS; ++c) mx = fmaxf(mx, logit[c]);
    float s = 0.0f;
#pragma unroll
    for (int c = 0; c < NCLASS; ++c) s += __expf(logit[c] - mx);
    float lse = __logf(s);
#pragma unroll
    for (int c = 0; c < NCLASS; ++c) {
        float lo = logit[c] - mx - lse;               // log_softmax
        out[g * NCLASS + c] = lo;
        out[NGRAPH * NCLASS + g * NCLASS + c] = __expf(lo);  // softmax
    }
}

// ---------------------------------------------------------------------------
extern "C" void kernel_launch(void* const* d_in, const int* in_sizes, int n_in,
                              void* d_out, int out_size, void* d_ws, size_t ws_size,
                              hipStream_t stream) {
    const float* norm = (const float*)d_in[0];
    const float* pos  = (const float*)d_in[1];
    const float* x    = (const float*)d_in[2];
    const long long* edge_index = (const long long*)d_in[3];
    const long long* batch      = (const long long*)d_in[4];
    const float* W1 = (const float*)d_in[5];
    const float* b1 = (const float*)d_in[6];
    const float* W2 = (const float*)d_in[7];
    const float* b2 = (const float*)d_in[8];
    const float* W3 = (const float*)d_in[9];
    const float* b3 = (const float*)d_in[10];
    const float* Wl = (const float*)d_in[11];
    const float* bl = (const float*)d_in[12];
    float* out = (float*)d_out;

    const long long* src = edge_index;          // row 0
    const long long* dst = edge_index + EDGES;  // row 1

    // Workspace layout (floats):
    //   H208 [N,208] = [h1(0:64) | inp(64:72) | h2(72:200) | inp(200:208)]
    //   M    [N,256] dense transform output per layer
    //   AGG  [N,256] layer-3 aggregation
    //   pooled [G,256] as uint bit patterns
    float* H   = (float*)d_ws;
    float* M   = H + (size_t)NODES * LDH;
    float* AGG = M + (size_t)NODES * 256;
    unsigned int* pooled = (unsigned int*)(AGG + (size_t)NODES * 256);

    const int T = 256;
    auto blks = [](long long n) { return (unsigned)((n + 255) / 256); };

    // inp -> H208 (both copies)
    pack_inp_kernel<<<blks((long long)NODES * 8), T, 0, stream>>>(norm, pos, x, H);
    zero_u32_kernel<<<blks(NGRAPH * 256), T, 0, stream>>>(pooled, NGRAPH * 256);

    // ---- Layer 1: [N,8] @ [8,64] -> scatter -> bias+ReLU into H cols 0..63
    gemm_wmma_kernel<8, LDH, 64><<<dim3(64 / 16, NODES / 16), 32, 0, stream>>>(H + 64, W1, M);
    zero_block_kernel<64><<<blks((long long)NODES * 64), T, 0, stream>>>(H, LDH, 0);
    scatter_add_kernel<64><<<blks((long long)EDGES * 64), T, 0, stream>>>(M, src, dst, H, LDH, 0);
    bias_relu_kernel<64><<<blks((long long)NODES * 64), T, 0, stream>>>(H, LDH, 0, b1);

    // ---- Layer 2: [N,72] @ [72,128] -> scatter -> bias+ReLU into H cols 72..199
    gemm_wmma_kernel<72, LDH, 128><<<dim3(128 / 16, NODES / 16), 32, 0, stream>>>(H, W2, M);
    zero_block_kernel<128><<<blks((long long)NODES * 128), T, 0, stream>>>(H, LDH, 72);
    scatter_add_kernel<128><<<blks((long long)EDGES * 128), T, 0, stream>>>(M, src, dst, H, LDH, 72);
    bias_relu_kernel<128><<<blks((long long)NODES * 128), T, 0, stream>>>(H, LDH, 72, b2);

    // ---- Layer 3: [N,208] @ [208,256] -> scatter into AGG -> bias+ReLU
    gemm_wmma_kernel<208, LDH, 256><<<dim3(256 / 16, NODES / 16), 32, 0, stream>>>(H, W3, M);
    zero_block_kernel<256><<<blks((long long)NODES * 256), T, 0, stream>>>(AGG, 256, 0);
    scatter_add_kernel<256><<<blks((long long)EDGES * 256), T, 0, stream>>>(M, src, dst, AGG, 256, 0);
    bias_relu_kernel<256><<<blks((long long)NODES * 256), T, 0, stream>>>(AGG, 256, 0, b3);

    // ---- Pool + head
    pool_max_kernel<<<blks((long long)NODES * 256), T, 0, stream>>>(AGG, batch, pooled);
    head_kernel<<<1, 32, 0, stream>>>(pooled, Wl, bl, out);

    (void)in_sizes; (void)n_in; (void)out_size; (void)ws_size;
}